// WaveCell_309237646054
// MI455X (gfx1250) — compile-verified
//
#include <hip/hip_runtime.h>
#include <cstdint>

// ---------------------------------------------------------------------------
// Problem constants (match reference)
// ---------------------------------------------------------------------------
constexpr int    NXg = 512, NYg = 512, BATCH = 4, Tn = 128;
constexpr int    N2  = NXg * NYg;
constexpr int    SRC_X = 30, SRC_Y = 256;
constexpr int    PRB_X = 482;                 // probe rows; cols 128/256/384
constexpr int    DRX0 = 60, DRX1 = 452, DRY0 = 60, DRY1 = 452;
constexpr int    RD   = DRX1 - DRX0;          // 392
constexpr float  NL_B0 = 0.1f;
constexpr float  ETA = 0.5f, BETA = 100.0f;
constexpr int    PML_N = 20;
constexpr float  PML_MAX = 3.0f, PML_P = 4.0f;
constexpr float  C0c = 1.0f, C1c = 0.9f;
constexpr double Hh   = 0.5 * 2.01;                    // = DT*2.01 in double
constexpr float  LAPC = (float)(1.0 / (Hh * Hh));      // h^-2, f32 like jnp
constexpr float  CEN  = -4.0f * LAPC;
constexpr float  INV_DT2 = 4.0f;                        // 1/DT^2, half_inv_dt=1

// Tile geometry: 16 rows x 64 cols per 256-thread block (8 x wave32).
// LDS row layout: [0..2 pad][3: west halo][4..67: interior][68: east halo]
// Row stride 72 floats = 288 B (multiple of 16) so interior cols are
// 16B-aligned in LDS -> b128 async transfers are legal on both ends.
constexpr int TX = 16, TY = 64;
constexpr int LW = TY + 8;                              // 72

// ---------------------------------------------------------------------------
// CDNA5 async global->LDS paths (guarded; graceful fallbacks)
// ---------------------------------------------------------------------------
#if defined(__has_builtin)
#if __has_builtin(__builtin_amdgcn_global_load_async_to_lds_b32) && \
    __has_builtin(__builtin_amdgcn_s_wait_asynccnt)
#define USE_ASYNC32 1
#if __has_builtin(__builtin_amdgcn_global_load_async_to_lds_b128)
#define USE_ASYNC128 1
#endif
#endif
#endif
#ifndef USE_ASYNC32
#define USE_ASYNC32 0
#endif
#ifndef USE_ASYNC128
#define USE_ASYNC128 0
#endif

#if USE_ASYNC32
typedef __attribute__((address_space(1))) int as1_int;   // global scalar
typedef __attribute__((address_space(3))) int as3_int;   // LDS scalar
#endif
#if USE_ASYNC128
typedef int v4i __attribute__((vector_size(4 * sizeof(int))));
typedef __attribute__((address_space(1))) v4i as1_v4i;   // global vector
typedef __attribute__((address_space(3))) v4i as3_v4i;   // LDS vector
#endif

// ---------------------------------------------------------------------------
// One-time setup: c2, b_bound, exp_rho, zero state + accumulators
// ---------------------------------------------------------------------------
__global__ __launch_bounds__(256) void wave_init(
    const float* __restrict__ rho,
    float* __restrict__ c2, float* __restrict__ bb, float* __restrict__ er,
    float* __restrict__ y1, float* __restrict__ y2, float* __restrict__ acc)
{
    const int idx = blockIdx.x * 256 + threadIdx.x;     // 0..N2-1
    const int ix = idx / NYg;
    const int iy = idx - ix * NYg;

    // exp_rho: LPF(rho) -> tanh projection inside design region, else 0
    float e = 0.0f;
    if (ix >= DRX0 && ix < DRX1 && iy >= DRY0 && iy < DRY1) {
        const int rx = ix - DRX0, ry = iy - DRY0;
        auto R = [&](int a, int b) -> float {
            return (a >= 0 && a < RD && b >= 0 && b < RD) ? rho[a * RD + b] : 0.0f;
        };
        const float lpf = 0.5f * R(rx, ry) +
                          0.125f * (R(rx - 1, ry) + R(rx + 1, ry) +
                                    R(rx, ry - 1) + R(rx, ry + 1));
        const float t_eta = tanhf(BETA * ETA);
        e = (t_eta + tanhf(BETA * (lpf - ETA))) /
            (t_eta + tanhf(BETA * (1.0f - ETA)));
    }
    er[idx] = e;
    const float c = C0c + (C1c - C0c) * e;
    c2[idx] = c * c;

    // b_bound: quartic PML ramps on all four edges, combined in quadrature
    float bx = 0.0f, by = 0.0f;
    if (ix <= PML_N)
        bx = PML_MAX * powf((float)(PML_N - ix) * (1.0f / PML_N), PML_P);
    else if (ix >= NXg - PML_N - 1)
        bx = PML_MAX * powf((float)(ix - (NXg - PML_N - 1)) * (1.0f / PML_N), PML_P);
    if (iy <= PML_N)
        by = PML_MAX * powf((float)(PML_N - iy) * (1.0f / PML_N), PML_P);
    else if (iy >= NYg - PML_N - 1)
        by = PML_MAX * powf((float)(iy - (NYg - PML_N - 1)) * (1.0f / PML_N), PML_P);
    bb[idx] = sqrtf(bx * bx + by * by);

    // zero wave state for all batches + probe accumulators
    #pragma unroll
    for (int b = 0; b < BATCH; ++b) {
        y1[(size_t)b * N2 + idx] = 0.0f;
        y2[(size_t)b * N2 + idx] = 0.0f;
    }
    if (idx < 16) acc[idx] = 0.0f;
}

// ---------------------------------------------------------------------------
// One FDTD time step.  grid = (NY/TY, NX/TX, BATCH), block = 256 (8 x wave32)
// ---------------------------------------------------------------------------
__global__ __launch_bounds__(256) void wave_step(
    const float* __restrict__ y1, const float* __restrict__ y2,
    float* __restrict__ yn,
    const float* __restrict__ c2, const float* __restrict__ bb,
    const float* __restrict__ er,
    const float* __restrict__ x, int t, float* __restrict__ acc)
{
    __shared__ float tile[(TX + 2) * LW];               // 18 x 72 floats

    const int tid = threadIdx.x;
    const int b   = blockIdx.z;
    const int x0  = blockIdx.y * TX;
    const int y0  = blockIdx.x * TY;
    const float* __restrict__ y1b = y1 + (size_t)b * N2;

    // ---- stage y1 halo tile (18 rows x 66 cols) into LDS ----
#if USE_ASYNC128
    // Interior: 18 rows x 16 quads, 16B-aligned on both ends -> b128 async.
    for (int e = tid; e < (TX + 2) * (TY / 4); e += 256) {   // 288 quads
        const int r  = e >> 4;
        const int qc = e & 15;
        const int gx = x0 - 1 + r;
        float* ldst = &tile[r * LW + 4 + qc * 4];
        if (gx >= 0 && gx < NXg) {
            const float* gp = y1b + gx * NYg + y0 + qc * 4;
            __builtin_amdgcn_global_load_async_to_lds_b128(
                (as1_v4i*)(uintptr_t)gp, (as3_v4i*)ldst, 0, 0);
        } else {
            float4 z; z.x = 0.0f; z.y = 0.0f; z.z = 0.0f; z.w = 0.0f;
            *reinterpret_cast<float4*>(ldst) = z;
        }
    }
    // West/east halo columns: 2 x 18 scalars -> b32 async.
    if (tid < 2 * (TX + 2)) {
        const bool left = tid < (TX + 2);
        const int r  = left ? tid : tid - (TX + 2);
        const int gx = x0 - 1 + r;
        const int gy = left ? (y0 - 1) : (y0 + TY);
        float* ldst = &tile[r * LW + (left ? 3 : 4 + TY)];
        if (gx >= 0 && gx < NXg && gy >= 0 && gy < NYg) {
            __builtin_amdgcn_global_load_async_to_lds_b32(
                (as1_int*)(uintptr_t)(y1b + gx * NYg + gy),
                (as3_int*)ldst, 0, 0);
        } else {
            *ldst = 0.0f;
        }
    }
#else
    for (int e = tid; e < (TX + 2) * (TY + 2); e += 256) {   // 1188 scalars
        const int r  = e / (TY + 2);
        const int c  = e - r * (TY + 2);
        const int gx = x0 - 1 + r;
        const int gy = y0 - 1 + c;
        float* ldst = &tile[r * LW + 3 + c];
        const bool inb = (gx >= 0) & (gx < NXg) & (gy >= 0) & (gy < NYg);
#if USE_ASYNC32
        if (inb) {
            __builtin_amdgcn_global_load_async_to_lds_b32(
                (as1_int*)(uintptr_t)(y1b + gx * NYg + gy),
                (as3_int*)ldst, 0, 0);
        } else {
            *ldst = 0.0f;
        }
#else
        *ldst = inb ? y1b[gx * NYg + gy] : 0.0f;
#endif
    }
#endif
#if USE_ASYNC32
    __builtin_amdgcn_s_wait_asynccnt(0);
#endif
    __syncthreads();

    // ---- 4 contiguous output points per thread ----
    const int r   = tid >> 4;            // 0..15
    const int q   = tid & 15;            // 0..15
    const int ix  = x0 + r;
    const int iy0 = y0 + q * 4;
    const size_t gbase = (size_t)b * N2 + (size_t)ix * NYg + iy0;
    const size_t cbase = (size_t)ix * NYg + iy0;

    const float4 y2v = *reinterpret_cast<const float4*>(y2 + gbase);
    const float4 c2v = *reinterpret_cast<const float4*>(c2 + cbase);
    const float4 bbv = *reinterpret_cast<const float4*>(bb + cbase);
    const float4 erv = *reinterpret_cast<const float4*>(er + cbase);
    const float* y2a = reinterpret_cast<const float*>(&y2v);
    const float* c2a = reinterpret_cast<const float*>(&c2v);
    const float* bba = reinterpret_cast<const float*>(&bbv);
    const float* era = reinterpret_cast<const float*>(&erv);

    const int lr = r + 1;
    const int lc = 4 + q * 4;            // interior starts at LDS col 4
    float o[4];
    #pragma unroll
    for (int j = 0; j < 4; ++j) {
        const float yc = tile[lr * LW + lc + j];
        const float nn = tile[(lr - 1) * LW + lc + j];
        const float ss = tile[(lr + 1) * LW + lc + j];
        const float ww = tile[lr * LW + lc + j - 1];
        const float ee = tile[lr * LW + lc + j + 1];
        const float lap = LAPC * (nn + ss + ww + ee) + CEN * yc;
        const float bq  = bba[j] + era[j] * (NL_B0 / (1.0f + yc * yc));
        const float num = 8.0f * yc - (INV_DT2 - bq) * y2a[j] + c2a[j] * lap;
        o[j] = (1.0f / (INV_DT2 + bq)) * num;       // match reference op order
    }

    // source injection (SRC_Y % 4 == 0 -> lane j = 0)
    if (ix == SRC_X && iy0 == SRC_Y) o[0] += x[b * Tn + t];

    float4 ov; ov.x = o[0]; ov.y = o[1]; ov.z = o[2]; ov.w = o[3];
    *reinterpret_cast<float4*>(yn + gbase) = ov;

    // probe intensity accumulation (unique writer per (batch,probe) per step;
    // steps are sequential kernels -> plain RMW is safe)
    if (ix == PRB_X) {
        if      (iy0 == 128) acc[b * 3 + 0] += o[0] * o[0];
        else if (iy0 == 256) acc[b * 3 + 1] += o[0] * o[0];
        else if (iy0 == 384) acc[b * 3 + 2] += o[0] * o[0];
    }
}

// ---------------------------------------------------------------------------
// Normalize: out[b,i] = I[b,i] / sum_i I[b,i]
// ---------------------------------------------------------------------------
__global__ void wave_finalize(const float* __restrict__ acc, float* __restrict__ out)
{
    const int i = threadIdx.x;
    if (i < BATCH * 3) {
        const int b = i / 3;
        const float s = acc[b * 3 + 0] + acc[b * 3 + 1] + acc[b * 3 + 2];
        out[i] = acc[i] / s;
    }
}

// ---------------------------------------------------------------------------
// Host driver (graph-capture safe: only kernel launches on `stream`)
// Workspace layout (floats): c2[N2] | bb[N2] | er[N2] | acc[64] | 3 x y[B*N2]
// Total ~15.8 MB.
// ---------------------------------------------------------------------------
extern "C" void kernel_launch(void* const* d_in, const int* in_sizes, int n_in,
                              void* d_out, int out_size, void* d_ws, size_t ws_size,
                              hipStream_t stream)
{
    (void)in_sizes; (void)n_in; (void)out_size; (void)ws_size;
    const float* x   = (const float*)d_in[0];   // (BATCH, T)
    const float* rho = (const float*)d_in[1];   // (392, 392)
    float* out = (float*)d_out;                 // (BATCH, 3)

    float* c2   = (float*)d_ws;
    float* bb   = c2  + N2;
    float* er   = bb  + N2;
    float* acc  = er  + N2;              // 12 used, 64 reserved (alignment)
    float* buf0 = acc + 64;
    float* buf1 = buf0 + (size_t)BATCH * N2;
    float* buf2 = buf1 + (size_t)BATCH * N2;

    wave_init<<<N2 / 256, 256, 0, stream>>>(rho, c2, bb, er, buf0, buf1, acc);

    float* yp   = buf0;   // y1 (current)
    float* ypp  = buf1;   // y2 (previous)
    float* ynew = buf2;
    for (int t = 0; t < Tn; ++t) {
        wave_step<<<dim3(NYg / TY, NXg / TX, BATCH), 256, 0, stream>>>(
            yp, ypp, ynew, c2, bb, er, x, t, acc);
        float* tmp = ypp; ypp = yp; yp = ynew; ynew = tmp;   // (y, y1) -> carry
    }

    wave_finalize<<<1, 32, 0, stream>>>(acc, out);
}